// TreeToTreeAttentionBatch_50465865728462
// MI455X (gfx1250) — compile-verified
//
#include <hip/hip_runtime.h>
#include <hip/hip_bf16.h>
#include <math.h>

// Dims from reference: B,N,H,E,NCLASS = 1024,256,256,256,512
#define B_    1024
#define N_    256
#define H_    256
#define E_    256
#define OUTW  (E_ + H_)          // 512 output columns
#define CH    64                 // rows of annotations staged per LDS chunk
#define CSTR  260                // padded LDS row stride in floats (1040B, 16B-aligned)

typedef __attribute__((ext_vector_type(2))) float v2f;
typedef __attribute__((ext_vector_type(8))) float v8f;
typedef int v4i __attribute__((vector_size(4 * sizeof(int))));

// ---------------- async global->LDS support (guarded probes) ----------------
#if defined(__has_builtin)
#  if __has_builtin(__builtin_amdgcn_global_load_async_to_lds_b128)
#    define HAVE_ASYNC_LDS 1
#  endif
#  if __has_builtin(__builtin_amdgcn_s_wait_asynccnt)
#    define HAVE_WAIT_ASYNC 1
#  endif
#endif

#if defined(HAVE_ASYNC_LDS)
__device__ __forceinline__ void async_copy_b128(const void* g, void* l) {
  __builtin_amdgcn_global_load_async_to_lds_b128((v4i*)g, (v4i*)l, 0, 0);
}
__device__ __forceinline__ void wait_async_16() {
#  if defined(HAVE_WAIT_ASYNC)
  __builtin_amdgcn_s_wait_asynccnt(16);
#  else
  asm volatile("s_wait_asynccnt 16" ::: "memory");
#  endif
}
__device__ __forceinline__ void wait_async_0() {
#  if defined(HAVE_WAIT_ASYNC)
  __builtin_amdgcn_s_wait_asynccnt(0);
#  else
  asm volatile("s_wait_asynccnt 0" ::: "memory");
#  endif
}
#endif

// ---------------------------------------------------------------------------
// Kernel A: V[b,h] = sum_o D[b,o] * W_ah[o,h]   (b_ah dropped: cancels in softmax)
// fp32 WMMA 16x16x4. One wave per 16x16 output tile; 8 waves/block.
// ---------------------------------------------------------------------------
__global__ __launch_bounds__(256) void kA_vgemm(const float* __restrict__ D,
                                                const float* __restrict__ W,
                                                float* __restrict__ V) {
  const int wave = blockIdx.x * 8 + (threadIdx.x >> 5);
  const int lane = threadIdx.x & 31;
  const int tb = wave >> 4;               // 64 tiles over B
  const int tn = wave & 15;               // 16 tiles over H
  const int b0 = tb * 16, n0 = tn * 16;
  const int half = lane >> 4;             // 0 | 1
  const int m    = lane & 15;
  const int koff = half * 2;
  const float* drow = D + (size_t)(b0 + m) * H_;
  v8f acc = {0.f,0.f,0.f,0.f,0.f,0.f,0.f,0.f};
  for (int k = 0; k < H_; k += 4) {
    v2f a, bf;
    a.x  = drow[k + koff];
    a.y  = drow[k + koff + 1];
    bf.x = W[(size_t)(k + koff)     * H_ + n0 + m];
    bf.y = W[(size_t)(k + koff + 1) * H_ + n0 + m];
    acc = __builtin_amdgcn_wmma_f32_16x16x4_f32(false, a, false, bf,
                                                (short)0, acc, false, false);
  }
  const int col = n0 + m;
  #pragma unroll
  for (int r = 0; r < 8; ++r)
    V[(size_t)(b0 + r + half * 8) * H_ + col] = acc[r];
}

// ---------------------------------------------------------------------------
// Kernel B: fused attention per batch element (flash-style single HBM pass).
// One block (8 waves) per b. Stage 64x256 fp32 chunks of annotations[b] into
// LDS (double buffered, async), online softmax over n, accumulate context.
// ctx_ws[b,h] = sum_n softmax(logits)[n] * annotations[b,n,h]
// ---------------------------------------------------------------------------
__global__ __launch_bounds__(256) void kB_attn(const float* __restrict__ annot,
                                               const float* __restrict__ Vws,
                                               float* __restrict__ ctx_ws) {
  extern __shared__ float smem[];
  float* v_s  = smem;                     // H_ floats (16B aligned)
  float* lg   = v_s + H_;                 // CH logits
  float* pb   = lg + CH;                  // CH softmax weights
  float* buf0 = pb + CH;                  // CH * CSTR (rows 16B aligned)
  float* buf1 = buf0 + CH * CSTR;         // CH * CSTR

  const int b   = blockIdx.x;
  const int tid = threadIdx.x;
  const float* Ab = annot + (size_t)b * N_ * H_;

  v_s[tid] = Vws[(size_t)b * H_ + tid];

  const int rr = tid >> 2, jj = tid & 3;  // 4 threads per staged row

  auto issue = [&](int c) {               // copy chunk c into buffer c&1
    float* dst = ((c & 1) ? buf1 : buf0) + rr * CSTR + jj * 64;
    const float* src = Ab + (size_t)(c * CH + rr) * H_ + jj * 64;
#if defined(HAVE_ASYNC_LDS)
    #pragma unroll
    for (int i = 0; i < 16; ++i) async_copy_b128(src + i * 4, dst + i * 4);
#else
    #pragma unroll
    for (int i = 0; i < 16; ++i)
      *(float4*)(dst + i * 4) = *(const float4*)(src + i * 4);
#endif
  };

  issue(0);
  float ctx = 0.f, m_run = -3.0e38f, l_run = 0.f;

  for (int c = 0; c < N_ / CH; ++c) {
    const bool more = (c + 1 < N_ / CH);
    if (more) issue(c + 1);
#if defined(HAVE_ASYNC_LDS)
    if (more) wait_async_16(); else wait_async_0();
#endif
    __syncthreads();                      // chunk c resident (and v_s on c==0)
    const float* buf = ((c & 1) ? buf1 : buf0);

    {                                     // logits: 4 threads/row, b128 LDS reads
      const float4* row4 = (const float4*)(buf + rr * CSTR) + jj * 16;
      const float4* v4   = (const float4*)v_s + jj * 16;
      float s = 0.f;
      #pragma unroll 4
      for (int i = 0; i < 16; ++i) {
        const float4 a4 = row4[i], b4 = v4[i];
        s = fmaf(a4.x, b4.x, s); s = fmaf(a4.y, b4.y, s);
        s = fmaf(a4.z, b4.z, s); s = fmaf(a4.w, b4.w, s);
      }
      s += __shfl_xor(s, 1, 32);          // reduce the 4 partials per row
      s += __shfl_xor(s, 2, 32);
      if (jj == 0) lg[rr] = s;
    }
    __syncthreads();

    float cmax = -3.0e38f;                // uniform across block (broadcast reads)
    for (int r = 0; r < CH; ++r) cmax = fmaxf(cmax, lg[r]);
    const float m_new = fmaxf(m_run, cmax);
    const float scale = __expf(m_run - m_new);
    if (tid < CH) pb[tid] = __expf(lg[tid] - m_new);
    __syncthreads();

    float lsum = 0.f, csum = 0.f;         // thread tid owns h = tid
    for (int r = 0; r < CH; ++r) {
      const float p = pb[r];
      lsum += p;
      csum = fmaf(p, buf[r * CSTR + tid], csum);
    }
    ctx   = ctx * scale + csum;
    l_run = l_run * scale + lsum;
    m_run = m_new;
    __syncthreads();                      // done reading buf before its reuse
  }
  ctx_ws[(size_t)b * H_ + tid] = ctx / l_run;
}

// ---------------------------------------------------------------------------
// Kernel C: et = tanh([dec, ctx] @ W_ps^T + b_ps), written to out[:, E_:].
// fp32 WMMA 16x16x4, K = 2H = 512 (first half from dec, second from ctx).
// ---------------------------------------------------------------------------
__global__ __launch_bounds__(256) void kC_presoftmax(const float* __restrict__ dec,
                                                     const float* __restrict__ ctxw,
                                                     const float* __restrict__ Wps,
                                                     const float* __restrict__ bps,
                                                     float* __restrict__ out) {
  const int wave = blockIdx.x * 8 + (threadIdx.x >> 5);
  const int lane = threadIdx.x & 31;
  const int tb = wave >> 4, tn = wave & 15;
  const int b0 = tb * 16, n0 = tn * 16;
  const int half = lane >> 4, m = lane & 15, koff = half * 2;
  const float* dl   = dec  + (size_t)(b0 + m) * H_;
  const float* cl   = ctxw + (size_t)(b0 + m) * H_;
  const float* wcol = Wps  + (size_t)(n0 + m) * (2 * H_);   // B[k][n] = W_ps[n][k]
  v8f acc = {0.f,0.f,0.f,0.f,0.f,0.f,0.f,0.f};
  for (int k = 0; k < 2 * H_; k += 4) {
    const int kk = k + koff;
    v2f a, bf;
    if (k < H_) { a.x = dl[kk];      a.y = dl[kk + 1]; }
    else        { a.x = cl[kk - H_]; a.y = cl[kk - H_ + 1]; }
    bf.x = wcol[kk];
    bf.y = wcol[kk + 1];
    acc = __builtin_amdgcn_wmma_f32_16x16x4_f32(false, a, false, bf,
                                                (short)0, acc, false, false);
  }
  const int col = n0 + m;
  const float bias = bps[col];
  #pragma unroll
  for (int r = 0; r < 8; ++r)
    out[(size_t)(b0 + r + half * 8) * OUTW + E_ + col] = tanhf(acc[r] + bias);
}

// Kernel D: out[:, 0:E] = emb[next_input]
__global__ __launch_bounds__(256) void kD_emb(const float* __restrict__ emb,
                                              const int* __restrict__ idx,
                                              float* __restrict__ out) {
  const int b = blockIdx.x, e = threadIdx.x;
  out[(size_t)b * OUTW + e] = emb[(size_t)idx[b] * E_ + e];
}

extern "C" void kernel_launch(void* const* d_in, const int* in_sizes, int n_in,
                              void* d_out, int out_size, void* d_ws, size_t ws_size,
                              hipStream_t stream) {
  (void)in_sizes; (void)n_in; (void)out_size; (void)ws_size;
  const float* dec   = (const float*)d_in[0];
  const float* annot = (const float*)d_in[1];
  const float* W_ah  = (const float*)d_in[2];
  // d_in[3] (b_ah) intentionally unused: per-row constant, cancels in softmax
  const float* W_ps  = (const float*)d_in[4];
  const float* b_ps  = (const float*)d_in[5];
  const float* emb   = (const float*)d_in[6];
  const int*   nxt   = (const int*)d_in[7];
  float* out = (float*)d_out;

  float* Vws   = (float*)d_ws;                 // B*H floats
  float* ctxws = Vws + (size_t)B_ * H_;        // B*H floats

  const int tiles = (B_ / 16) * (H_ / 16);     // 1024 16x16 tiles, 8 waves/block
  kA_vgemm<<<dim3(tiles / 8), dim3(256), 0, stream>>>(dec, W_ah, Vws);

  const size_t smem = (size_t)(H_ + 2 * CH + 2 * CH * CSTR) * sizeof(float); // ~131.5 KB
  kB_attn<<<dim3(B_), dim3(256), smem, stream>>>(annot, Vws, ctxws);

  kC_presoftmax<<<dim3(tiles / 8), dim3(256), 0, stream>>>(dec, ctxws, W_ps, b_ps, out);
  kD_emb<<<dim3(B_), dim3(E_), 0, stream>>>(emb, nxt, out);
}